// GraphSAGENodeClassifier_43490838839925
// MI455X (gfx1250) — compile-verified
//
#include <hip/hip_runtime.h>

// GraphSAGE 3-layer forward for MI455X (gfx1250).
// fp32 exact path -> V_WMMA_F32_16X16X4_F32 for the two 100k x 128 x 64 GEMMs.
// Edge aggregation via global fp32 atomics (L2-resident working set).

#define NN 100000      // N_NODES
#define NE 1200000     // N_EDGES
#define D  64          // IN_CH == HID

typedef __attribute__((ext_vector_type(2))) float v2f;
typedef __attribute__((ext_vector_type(8))) float v8f;

// ---------------- degree / inv_deg ----------------
__global__ void deg_kernel(const int* __restrict__ dst, float* __restrict__ deg) {
    int e = blockIdx.x * blockDim.x + threadIdx.x;
    if (e < NE) atomicAdd(&deg[dst[e]], 1.0f);
}

__global__ void invdeg_kernel(float* __restrict__ deg) {
    int i = blockIdx.x * blockDim.x + threadIdx.x;
    if (i < NN) deg[i] = 1.0f / fmaxf(deg[i], 1.0f);
}

// ---------------- edge scatter-add: agg[dst] += h[src] ----------------
// One 32-thread group per edge; each thread moves 2 channels (float2).
__global__ __launch_bounds__(256) void scatter_kernel(
    const float* __restrict__ h, const int* __restrict__ src,
    const int* __restrict__ dst, float* __restrict__ agg)
{
    int gid  = blockIdx.x * 256 + threadIdx.x;
    int e    = gid >> 5;
    int lane = gid & 31;
    if (e >= NE) return;
    int s = src[e];
    int d = dst[e];
    const float2* hp = (const float2*)(h + (size_t)s * D);
    float2 v = hp[lane];
    float* ap = agg + (size_t)d * D + 2 * lane;
    atomicAdd(ap,     v.x);
    atomicAdd(ap + 1, v.y);
}

// ---------------- fused SAGE GEMM:  out = relu( [agg*inv, hin] @ [Wl;Wr]^T + b ) ---------
// Block: 256 threads = 8 waves; covers 32 nodes x 64 out-channels.
// Wave (w) -> row tile (w/4)*16, col tile (w%4)*16. K = 128 (agg part then root part).
// A frag (16x4 f32): lane holds {A[row][k+2hi], A[row][k+2hi+1]},  row = lane%16, hi = lane/16
// B frag (4x16 f32): lane holds {B[k+2hi][col], B[k+2hi+1][col]}, col = lane%16
// C/D: VGPR i -> M = i + 8*hi, N = lane%16.
#define AS 40   // LDS row stride for A tiles: 2*AS mod 64 == 16 -> lo/hi banks disjoint
#define BS 72   // LDS row stride for B tiles: 2*BS mod 64 == 16 -> lo/hi banks disjoint

__global__ __launch_bounds__(256) void sage_gemm_kernel(
    const float* __restrict__ agg, const float* __restrict__ invdeg,
    const float* __restrict__ hin,
    const float* __restrict__ Wl, const float* __restrict__ Wr,
    const float* __restrict__ bias, float* __restrict__ out, int relu)
{
    __shared__ float As[128 * AS];   // A^T tile: As[k][rowLocal], 32 rows
    __shared__ float Bs[128 * BS];   // B tile  : Bs[k][n], n = out channel 0..63

    const int t       = threadIdx.x;
    const int rowBase = blockIdx.x * 32;

    // Stage A: concat( agg*inv_deg , hin ) for 32 nodes -> As[k][r]  (coalesced global reads)
    #pragma unroll
    for (int i = 0; i < 16; ++i) {
        int e = i * 256 + t;
        int r = e >> 7;            // 0..31
        int k = e & 127;           // 0..127
        int node = rowBase + r;
        float v;
        if (k < D) v = agg[(size_t)node * D + k] * invdeg[node];
        else       v = hin[(size_t)node * D + (k - D)];
        As[k * AS + r] = v;
    }
    // Stage B: [Wl^T ; Wr^T] as Bs[k][n] (k = input ch 0..127, n = out ch 0..63)
    #pragma unroll
    for (int i = 0; i < 32; ++i) {
        int e = i * 256 + t;
        int n = e >> 7;            // 0..63
        int k = e & 127;
        float w = (k < D) ? Wl[n * D + k] : Wr[n * D + (k - D)];
        Bs[k * BS + n] = w;
    }
    __syncthreads();

    const int wave = t >> 5;
    const int lane = t & 31;
    const int hi   = lane >> 4;    // 0: K pair {0,1}, 1: K pair {2,3}
    const int l16  = lane & 15;
    const int tRow = (wave >> 2) * 16;   // 0 / 16
    const int tCol = (wave & 3) * 16;    // 0 / 16 / 32 / 48

    // bias pre-loaded: every acc VGPR holds N = l16 -> same bias value
    v8f acc;
    {
        float bv = bias[tCol + l16];
        #pragma unroll
        for (int i = 0; i < 8; ++i) acc[i] = bv;
    }

    const int aRow = tRow + l16;
    const int bCol = tCol + l16;

    #pragma unroll
    for (int kk = 0; kk < 128; kk += 4) {
        int kA = kk + 2 * hi;
        v2f a, b;
        a.x = As[kA * AS + aRow];
        a.y = As[(kA + 1) * AS + aRow];
        b.x = Bs[kA * BS + bCol];
        b.y = Bs[(kA + 1) * BS + bCol];
        acc = __builtin_amdgcn_wmma_f32_16x16x4_f32(
            /*neg_a=*/false, a, /*neg_b=*/false, b,
            /*c_mod=*/(short)0, acc, /*reuse_a=*/false, /*reuse_b=*/false);
    }

    #pragma unroll
    for (int i = 0; i < 8; ++i) {
        float v = acc[i];
        if (relu) v = fmaxf(v, 0.0f);
        int m = i + 8 * hi;
        out[(size_t)(rowBase + tRow + m) * D + tCol + l16] = v;
    }
}

// ---------------- layer 3: OUT_CH = 2 (0.05 GFLOP -> scalar) ----------------
__global__ __launch_bounds__(256) void out_kernel(
    const float* __restrict__ agg, const float* __restrict__ invdeg,
    const float* __restrict__ hin,
    const float* __restrict__ Wl, const float* __restrict__ Wr,
    const float* __restrict__ bias, float* __restrict__ out)
{
    int n = blockIdx.x * blockDim.x + threadIdx.x;
    if (n >= NN) return;
    float inv = invdeg[n];
    const float* ap = agg + (size_t)n * D;
    const float* hp = hin + (size_t)n * D;
    float a0 = 0.0f, a1 = 0.0f;
    #pragma unroll 8
    for (int k = 0; k < D; ++k) {
        float av = ap[k] * inv;
        float hv = hp[k];
        a0 += av * Wl[k]     + hv * Wr[k];
        a1 += av * Wl[D + k] + hv * Wr[D + k];
    }
    out[(size_t)n * 2 + 0] = a0 + bias[0];
    out[(size_t)n * 2 + 1] = a1 + bias[1];
}

// ---------------- host orchestration ----------------
extern "C" void kernel_launch(void* const* d_in, const int* in_sizes, int n_in,
                              void* d_out, int out_size, void* d_ws, size_t ws_size,
                              hipStream_t stream)
{
    const float* x   = (const float*)d_in[0];
    const int*   ei  = (const int*)  d_in[1];   // [2, NE] int32
    const float* Wl1 = (const float*)d_in[2];
    const float* b1  = (const float*)d_in[3];
    const float* Wr1 = (const float*)d_in[4];
    const float* Wl2 = (const float*)d_in[5];
    const float* b2  = (const float*)d_in[6];
    const float* Wr2 = (const float*)d_in[7];
    const float* Wl3 = (const float*)d_in[8];
    const float* b3  = (const float*)d_in[9];
    const float* Wr3 = (const float*)d_in[10];
    float* out = (float*)d_out;

    const int* src = ei;
    const int* dst = ei + NE;

    // workspace layout (~77.2 MB): deg | agg | h1 | h2
    char* ws = (char*)d_ws;
    size_t o = 0;
    float* deg = (float*)(ws + o); o += ((size_t)NN * 4 + 255) & ~(size_t)255;
    float* agg = (float*)(ws + o); o += (size_t)NN * D * 4;
    float* h1  = (float*)(ws + o); o += (size_t)NN * D * 4;
    float* h2  = (float*)(ws + o); o += (size_t)NN * D * 4;
    (void)ws_size; (void)in_sizes; (void)n_in; (void)out_size;

    const int EB  = (NE + 255) / 256;        // edge-parallel blocks
    const int NB  = (NN + 255) / 256;        // node-parallel blocks
    const int SB  = (NE * 32) / 256;         // scatter blocks (exact: 150000)
    const int GB  = NN / 32;                 // gemm blocks (exact: 3125)

    // degrees -> inv_deg (in place)
    hipMemsetAsync(deg, 0, (size_t)NN * 4, stream);
    deg_kernel<<<EB, 256, 0, stream>>>(dst, deg);
    invdeg_kernel<<<NB, 256, 0, stream>>>(deg);

    // layer 1
    hipMemsetAsync(agg, 0, (size_t)NN * D * 4, stream);
    scatter_kernel<<<SB, 256, 0, stream>>>(x, src, dst, agg);
    sage_gemm_kernel<<<GB, 256, 0, stream>>>(agg, deg, x, Wl1, Wr1, b1, h1, 1);

    // layer 2
    hipMemsetAsync(agg, 0, (size_t)NN * D * 4, stream);
    scatter_kernel<<<SB, 256, 0, stream>>>(h1, src, dst, agg);
    sage_gemm_kernel<<<GB, 256, 0, stream>>>(agg, deg, h1, Wl2, Wr2, b2, h2, 1);

    // layer 3
    hipMemsetAsync(agg, 0, (size_t)NN * D * 4, stream);
    scatter_kernel<<<SB, 256, 0, stream>>>(h2, src, dst, agg);
    out_kernel<<<NB, 256, 0, stream>>>(agg, deg, h2, Wl3, Wr3, b3, out);
}